// MultiHeadAttention_57990648430719
// MI455X (gfx1250) — compile-verified
//
#include <hip/hip_runtime.h>
#include <hip/hip_bf16.h>
#include <math.h>

// ---------------- problem constants ----------------
constexpr int BATCH = 4;
constexpr int SEQ   = 2048;
constexpr int DMODEL = 512;
constexpr int DK     = 64;            // k_size == v_size
constexpr int MROWS  = BATCH * SEQ;   // 8192 flattened (b,l) rows
// reference uses temperature = sqrt(MODEL_SIZE)
constexpr float INV_TEMP = 0.044194173824159216f;  // 1/sqrt(512)

typedef __attribute__((ext_vector_type(2))) float v2f;
typedef __attribute__((ext_vector_type(4))) float v4f;
typedef __attribute__((ext_vector_type(8))) float v8f;
typedef int v4i_vs __attribute__((vector_size(16)));   // matches builtin proto

// fp32 WMMA: D(16x16) = A(16x4) * B(4x16) + C   (wave32, 2 VGPRs A/B, 8 VGPRs C/D)
static __device__ __forceinline__ v8f wmma_f32x4(v2f a, v2f b, v8f c) {
  return __builtin_amdgcn_wmma_f32_16x16x4_f32(false, a, false, b, (short)0, c,
                                               false, false);
}

// ---- CDNA5 async global->LDS copy (ASYNCcnt), with synchronous fallback ----
#if defined(__has_builtin)
#if __has_builtin(__builtin_amdgcn_global_load_async_to_lds_b128) && \
    __has_builtin(__builtin_amdgcn_s_wait_asynccnt)
#define HAVE_ASYNC_LDS 1
#endif
#endif

static __device__ __forceinline__ void copy128_to_lds(const float* g, float* l) {
#ifdef HAVE_ASYNC_LDS
  __builtin_amdgcn_global_load_async_to_lds_b128(
      (v4i_vs*)g, (v4i_vs*)l, /*offset=*/0, /*cpol=*/0);
#else
  *(v4f*)l = *(const v4f*)g;
#endif
}
static __device__ __forceinline__ void wait_lds_stage() {
#ifdef HAVE_ASYNC_LDS
  __builtin_amdgcn_s_wait_asynccnt(0);
#endif
}

// ---------------------------------------------------------------------------
// Kernel 0: fold Wp (512 x 512, heads tiled) into WpEff (512 x 64)
// WpEff[m][d] = sum_{r<8} Wp[m*512 + r*64 + d]
// ---------------------------------------------------------------------------
__global__ __launch_bounds__(256) void fold_wp_kernel(
    const float* __restrict__ Wp, float* __restrict__ WpEff) {
  int idx = blockIdx.x * 256 + threadIdx.x;   // 0 .. 512*64-1
  int m = idx >> 6;
  int d = idx & 63;
  float s = 0.f;
#pragma unroll
  for (int r = 0; r < 8; ++r) s += Wp[m * 512 + r * 64 + d];
  WpEff[idx] = s;
}

// ---------------------------------------------------------------------------
// Kernel 1: generic GEMM  out[M x N] = X[M x K] * W[N x K]^T + bias[N]
// grid = (M/64, N/64), block = 128 (4 waves). Wave w: rows [64*bx + 16w, +16),
// all 64 cols of the block's N-tile. W tile staged in LDS in 64-col K-chunks
// via async global->LDS copies.
// ---------------------------------------------------------------------------
#define WS_STRIDE 68   // 68 mod 64 = 4 -> conflict-free b64 B-fragment loads
__global__ __launch_bounds__(128) void gemm_xwt_kernel(
    const float* __restrict__ X, const float* __restrict__ W,
    const float* __restrict__ bias, float* __restrict__ out,
    int M, int N, int K) {
  __shared__ float Ws[64 * WS_STRIDE];

  const int tid   = threadIdx.x;
  const int wave  = tid >> 5;
  const int lane  = tid & 31;
  const int half  = lane >> 4;      // 0: lanes 0-15, 1: lanes 16-31
  const int l16   = lane & 15;
  const int rowBase = blockIdx.x * 64 + wave * 16;
  const int nBase   = blockIdx.y * 64;

  v8f acc[4];
#pragma unroll
  for (int j = 0; j < 4; ++j)
#pragma unroll
    for (int r = 0; r < 8; ++r) acc[j][r] = 0.0f;

  const float* xrow = X + (size_t)(rowBase + l16) * K;

  for (int kc = 0; kc < K; kc += 64) {
    __syncthreads();
    // stage W[nBase..+64)[kc..kc+64) -> Ws  (1024 x b128 over 128 threads)
#pragma unroll
    for (int i = 0; i < 8; ++i) {
      int idx = i * 128 + tid;        // float4 index
      int r  = idx >> 4;              // 0..63 row of tile
      int c4 = idx & 15;              // float4 within row
      copy128_to_lds(&W[(size_t)(nBase + r) * K + kc + c4 * 4],
                     &Ws[r * WS_STRIDE + c4 * 4]);
    }
    wait_lds_stage();
    __syncthreads();

    if (kc + 64 < K)
      __builtin_prefetch(&xrow[kc + 64], 0, 1);   // global_prefetch_b8

#pragma unroll
    for (int k0 = 0; k0 < 64; k0 += 4) {
      // A fragment: X[rowBase + l16][kc+k0 + 2*half + {0,1}]
      v2f a = *(const v2f*)&xrow[kc + k0 + 2 * half];
#pragma unroll
      for (int j = 0; j < 4; ++j) {
        // B fragment: W[nBase+16j+l16][kc+k0 + 2*half + {0,1}]
        v2f b = *(const v2f*)&Ws[(j * 16 + l16) * WS_STRIDE + k0 + 2 * half];
        acc[j] = wmma_f32x4(a, b, acc[j]);
      }
    }
  }

  // epilogue: C layout -> lane holds col n=l16, rows r + 8*half
#pragma unroll
  for (int j = 0; j < 4; ++j) {
    int n = nBase + j * 16 + l16;
    float bv = bias[n];
#pragma unroll
    for (int r = 0; r < 8; ++r) {
      int m = rowBase + r + 8 * half;
      out[(size_t)m * N + n] = acc[j][r] + bv;
    }
  }
}

// ---------------------------------------------------------------------------
// Kernel 2: flash attention over projected QP/KP/VP (each MROWS x 64).
// grid = (SEQ/64, BATCH), block = 128 (4 waves). Wave w owns 16 queries.
// Keys processed in chunks of 32, double-buffered in LDS with async copies:
// chunk i+1 is in flight (ASYNCcnt) while chunk i is consumed.
// ---------------------------------------------------------------------------
#define KP_STRIDE 68   // b64 reads along k: bank stride 4 -> conflict free
#define VP_STRIDE 72   // scalar reads along n, rows +2 apart: halves disjoint
#define PT_STRIDE 18   // even -> aligned b64 A-fragment reloads
#define KCHUNK 32
__global__ __launch_bounds__(128) void flash_attn_kernel(
    const float* __restrict__ QP, const float* __restrict__ KP,
    const float* __restrict__ VP, const unsigned char* __restrict__ mask,
    float* __restrict__ AO) {
  __shared__ float KPs[2][KCHUNK * KP_STRIDE];
  __shared__ float VPs[2][KCHUNK * VP_STRIDE];
  __shared__ float PtAll[4 * 16 * PT_STRIDE];

  const int tid  = threadIdx.x;
  const int wave = tid >> 5;
  const int lane = tid & 31;
  const int half = lane >> 4;
  const int l16  = lane & 15;
  const int bb   = blockIdx.y;
  const int qBase = blockIdx.x * 64 + wave * 16;   // query base for this wave

  float* Pt = &PtAll[wave * 16 * PT_STRIDE];

  // Preload A fragments of QP tile: aq[t] covers k-dim [4t, 4t+4)
  v2f aq[16];
  {
    const float* qrow = QP + ((size_t)(bb * SEQ + qBase + l16)) * DK;
#pragma unroll
    for (int t = 0; t < 16; ++t) aq[t] = *(const v2f*)&qrow[4 * t + 2 * half];
  }

  // online softmax state, lane-local for rows {r + 8*half}
  float mx[8], lsum[8];
  v8f oacc[4];
#pragma unroll
  for (int r = 0; r < 8; ++r) { mx[r] = -3.0e38f; lsum[r] = 0.0f; }
#pragma unroll
  for (int j = 0; j < 4; ++j)
#pragma unroll
    for (int r = 0; r < 8; ++r) oacc[j][r] = 0.0f;

  const unsigned char* mbase =
      mask + ((size_t)(bb * SEQ + qBase + 8 * half)) * SEQ + l16;

  // stage one 32-key chunk of KP/VP into LDS buffer `buf` (1024 b128 total)
  auto stage = [&](int buf, int kChunk) {
#pragma unroll
    for (int i = 0; i < 8; ++i) {
      int idx = i * 128 + tid;     // float4 index, 0..1023
      int sub = idx & 511;
      int r  = sub >> 4;           // 0..31
      int c4 = sub & 15;
      size_t goff = ((size_t)(bb * SEQ + kChunk + r)) * DK + c4 * 4;
      if (idx < 512)
        copy128_to_lds(&KP[goff], &KPs[buf][r * KP_STRIDE + c4 * 4]);
      else
        copy128_to_lds(&VP[goff], &VPs[buf][r * VP_STRIDE + c4 * 4]);
    }
  };

  constexpr int NCHUNK = SEQ / KCHUNK;   // 64
  stage(0, 0);                            // prologue: chunk 0 -> buffer 0

  for (int ic = 0; ic < NCHUNK; ++ic) {
    const int cur = ic & 1;
    wait_lds_stage();     // our async loads for buffer `cur` complete
    __syncthreads();      // everyone's loads complete; prior reads retired
    if (ic + 1 < NCHUNK) stage(cur ^ 1, (ic + 1) * KCHUNK);   // overlap

    const float* KPb = KPs[cur];
    const float* VPb = VPs[cur];

#pragma unroll 1
    for (int kt = 0; kt < KCHUNK / 16; ++kt) {
      const int kBase = ic * KCHUNK + kt * 16;

      // ---- S = QP_tile (16x64) * KP_tile^T (64x16) ----
      v8f c;
#pragma unroll
      for (int r = 0; r < 8; ++r) c[r] = 0.0f;
#pragma unroll
      for (int t = 0; t < 16; ++t) {
        v2f b = *(const v2f*)&KPb[(kt * 16 + l16) * KP_STRIDE + 4 * t + 2 * half];
        c = wmma_f32x4(aq[t], b, c);
      }

      // ---- mask + online softmax (rows r+8*half are lane-local) ----
      float s[8];
#pragma unroll
      for (int r = 0; r < 8; ++r) {
        unsigned char mb = mbase[(size_t)r * SEQ + kBase];
        float v = c[r] * INV_TEMP;
        s[r] = mb ? -INFINITY : v;
      }
#pragma unroll
      for (int r = 0; r < 8; ++r) {
        float tmax = s[r];
#pragma unroll
        for (int off = 1; off < 16; off <<= 1)
          tmax = fmaxf(tmax, __shfl_xor(tmax, off, 32));
        float mnew  = fmaxf(mx[r], tmax);
        float alpha = __expf(mx[r] - mnew);
        float p     = __expf(s[r] - mnew);
        mx[r] = mnew;
        float ps = p;
#pragma unroll
        for (int off = 1; off < 16; off <<= 1)
          ps += __shfl_xor(ps, off, 32);
        lsum[r] = lsum[r] * alpha + ps;
#pragma unroll
        for (int j = 0; j < 4; ++j) oacc[j][r] *= alpha;
        // P element (row r+8*half, col l16) -> per-wave LDS transpose buffer
        Pt[(r + 8 * half) * PT_STRIDE + l16] = p;
      }
      __builtin_amdgcn_wave_barrier();   // order LDS write -> read (same wave)

      // ---- O += P (16x16) * VP_tile (16x64) ----
#pragma unroll
      for (int cc = 0; cc < 4; ++cc) {
        // A fragment of P: row l16, k = 4cc + 2*half + {0,1}
        v2f pa = *(const v2f*)&Pt[l16 * PT_STRIDE + 4 * cc + 2 * half];
#pragma unroll
        for (int j = 0; j < 4; ++j) {
          int krow = kt * 16 + 4 * cc + 2 * half;
          v2f b;
          b.x = VPb[(krow + 0) * VP_STRIDE + j * 16 + l16];
          b.y = VPb[(krow + 1) * VP_STRIDE + j * 16 + l16];
          oacc[j] = wmma_f32x4(pa, b, oacc[j]);
        }
      }
      __builtin_amdgcn_wave_barrier();   // Pt reused next kt
    }
  }

  // ---- finalize: AO[b,l,:] = O / l ----
#pragma unroll
  for (int r = 0; r < 8; ++r) {
    float inv = 1.0f / lsum[r];
    size_t row = (size_t)(bb * SEQ + qBase + r + 8 * half) * DK;
#pragma unroll
    for (int j = 0; j < 4; ++j)
      AO[row + j * 16 + l16] = oacc[j][r] * inv;
  }
}

// ---------------------------------------------------------------------------
extern "C" void kernel_launch(void* const* d_in, const int* in_sizes, int n_in,
                              void* d_out, int out_size, void* d_ws, size_t ws_size,
                              hipStream_t stream) {
  const float* q  = (const float*)d_in[0];
  const float* k  = (const float*)d_in[1];
  const float* v  = (const float*)d_in[2];
  const unsigned char* attn_mask = (const unsigned char*)d_in[3];  // jnp.bool_
  const float* Wq = (const float*)d_in[4];
  const float* bq = (const float*)d_in[5];
  const float* Wk = (const float*)d_in[6];
  const float* bk = (const float*)d_in[7];
  const float* Wv = (const float*)d_in[8];
  const float* bv = (const float*)d_in[9];
  const float* Wp = (const float*)d_in[10];
  const float* bp = (const float*)d_in[11];
  float* out = (float*)d_out;

  // workspace layout (floats)
  float* QP    = (float*)d_ws;            // MROWS x 64
  float* KPp   = QP  + (size_t)MROWS * DK;
  float* VPp   = KPp + (size_t)MROWS * DK;
  float* AO    = VPp + (size_t)MROWS * DK;
  float* WpEff = AO  + (size_t)MROWS * DK;  // 512 x 64

  // 0) fold Wp over the 8 identical heads
  fold_wp_kernel<<<dim3((DMODEL * DK) / 256), dim3(256), 0, stream>>>(Wp, WpEff);

  // 1) projections: (8192 x 512) * (64 x 512)^T -> (8192 x 64)
  dim3 gProj(MROWS / 64, DK / 64);   // (128, 1)
  gemm_xwt_kernel<<<gProj, dim3(128), 0, stream>>>(q, Wq, bq, QP, MROWS, DK, DMODEL);
  gemm_xwt_kernel<<<gProj, dim3(128), 0, stream>>>(k, Wk, bk, KPp, MROWS, DK, DMODEL);
  gemm_xwt_kernel<<<gProj, dim3(128), 0, stream>>>(v, Wv, bv, VPp, MROWS, DK, DMODEL);

  // 2) flash attention -> AO (MROWS x 64)
  flash_attn_kernel<<<dim3(SEQ / 64, BATCH), dim3(128), 0, stream>>>(
      QP, KPp, VPp, attn_mask, AO);

  // 3) output projection: (8192 x 64) * (512 x 64)^T + bp -> (8192 x 512)
  dim3 gOut(MROWS / 64, DMODEL / 64);  // (128, 8)
  gemm_xwt_kernel<<<gOut, dim3(128), 0, stream>>>(AO, WpEff, bp, out, MROWS, DMODEL, DK);
}